// GraphConvolution_80427557585491
// MI455X (gfx1250) — compile-verified
//
#include <hip/hip_runtime.h>
#include <hip/hip_bf16.h>

// out = adj @ (X @ W) + bias   (all f32)
// N=8192, F_IN=256, F_OUT=128
//
// Unified f32 WMMA GEMM:  C[M x 128] = A[M x K] @ B[K x 128]  (+ bias init)
//   - wave32, V_WMMA_F32_16X16X4_F32 (16x16 tile, K-step 4)
//   - workgroup: 256 threads = 8 waves -> 128 rows; wave -> 16 rows x 128 cols
//   - B staged per 64-wide K tile into LDS, row-pair interleaved so each
//     B fragment pair is one conflict-free ds_load (compiler emits
//     ds_load_2addr_b64).
//   - Double-buffered LDS: next tile's global loads are issued before the
//     current tile's compute, spilled to the alternate buffer afterwards;
//     one barrier per tile, no exposed global-latency bubble.

typedef float v2f __attribute__((ext_vector_type(2)));
typedef float v4f __attribute__((ext_vector_type(4)));
typedef float v8f __attribute__((ext_vector_type(8)));

#define NODES   8192
#define FOUT    128
#define KT      64            // K tile staged in LDS
#define QS      288           // LDS float-stride per row-pair (256 data + 32 pad)
#define QTILE   ((KT / 2) * QS)   // floats per buffer (36,864 B)

__global__ __launch_bounds__(256) void gcn_wmma_gemm(
    const float* __restrict__ A, int lda,          // A[M x K], row stride lda
    const float* __restrict__ B, int K,            // B[K x 128] row-major
    const float* __restrict__ bias,                // nullptr -> init 0
    float* __restrict__ out)                       // out[M x 128]
{
    __shared__ float sB[2 * QTILE];                // 73,728 B LDS

    const int tid     = threadIdx.x;
    const int wave    = tid >> 5;
    const int lane    = tid & 31;
    const int lane_lo = lane & 15;   // M (for A) / N (for B,C) within tile
    const int lane_hi = lane >> 4;   // K-half selector
    const int row0    = blockIdx.x * 128 + wave * 16;

    // staging coordinates for this thread (8 float4 chunks per tile)
    const int s_r0  = tid >> 5;      // rows r0, r0+8, r0+16, ... (tid+i*256)>>5
    const int s_c4  = tid & 31;      // float4 column (constant across chunks)

    // ---- accumulators: 8 n-blocks of 16 cols, init with bias (or 0) ----
    v8f acc[8];
#pragma unroll
    for (int nb = 0; nb < 8; ++nb) {
        float b = bias ? bias[nb * 16 + lane_lo] : 0.0f;
        acc[nb] = (v8f){b, b, b, b, b, b, b, b};
    }

    // A fragment base: lane holds A[row0 + lane_lo][k + 2*lane_hi + {0,1}]
    const float* a_row = A + (size_t)(row0 + lane_lo) * (size_t)lda + 2 * lane_hi;

    // ---- prologue: stage tile 0 into buffer 0 ----
    v4f stg[8];
#pragma unroll
    for (int i = 0; i < 8; ++i) {
        const int r = s_r0 + i * 8;
        stg[i] = *(const v4f*)(B + (size_t)r * FOUT + s_c4 * 4);
    }
#pragma unroll
    for (int i = 0; i < 8; ++i) {
        const int r = s_r0 + i * 8;
        float* dst = &sB[(r >> 1) * QS + s_c4 * 8 + (r & 1)];
        dst[0] = stg[i].x; dst[2] = stg[i].y; dst[4] = stg[i].z; dst[6] = stg[i].w;
    }
    __syncthreads();

    int buf = 0;
    for (int k0 = 0; k0 < K; k0 += KT) {
        const bool has_next = (k0 + KT) < K;

        // ---- issue next tile's global loads (hidden behind compute) ----
        if (has_next) {
#pragma unroll
            for (int i = 0; i < 8; ++i) {
                const int r = s_r0 + i * 8;
                stg[i] = *(const v4f*)(B + (size_t)(k0 + KT + r) * FOUT + s_c4 * 4);
            }
            __builtin_prefetch((const void*)(a_row + k0 + KT), 0, 1);
        }

        // ---- compute: 16 k-steps of 4, 8 WMMAs each ----
        const float* a_ptr = a_row + k0;
        const float* sbuf  = &sB[buf * QTILE];
#pragma unroll 4
        for (int kk = 0; kk < KT; kk += 4) {
            const v2f afrag = *(const v2f*)(a_ptr + kk);        // global_load_b64
            const int q = (kk >> 1) + lane_hi;                  // row pair in LDS
#pragma unroll
            for (int nb = 0; nb < 8; ++nb) {
                // {B[kk+2hi][n], B[kk+2hi+1][n]} in one 8-byte LDS read
                const v2f bfrag =
                    *(const v2f*)&sbuf[q * QS + 2 * (nb * 16 + lane_lo)];
                acc[nb] = __builtin_amdgcn_wmma_f32_16x16x4_f32(
                    false, afrag, false, bfrag,
                    (short)0, acc[nb], false, false);
            }
        }

        // ---- spill staged registers into the alternate buffer ----
        if (has_next) {
            float* dbuf = &sB[(buf ^ 1) * QTILE];
#pragma unroll
            for (int i = 0; i < 8; ++i) {
                const int r = s_r0 + i * 8;
                float* dst = &dbuf[(r >> 1) * QS + s_c4 * 8 + (r & 1)];
                dst[0] = stg[i].x; dst[2] = stg[i].y;
                dst[4] = stg[i].z; dst[6] = stg[i].w;
            }
        }
        buf ^= 1;
        __syncthreads();
    }

    // ---- store: C layout vgpr v, lanes 0-15 -> M=v, lanes 16-31 -> M=v+8 ----
#pragma unroll
    for (int nb = 0; nb < 8; ++nb) {
        float* o = out + (size_t)(row0 + 8 * lane_hi) * FOUT + nb * 16 + lane_lo;
#pragma unroll
        for (int v = 0; v < 8; ++v)
            o[(size_t)v * FOUT] = acc[nb][v];
    }
}

extern "C" void kernel_launch(void* const* d_in, const int* in_sizes, int n_in,
                              void* d_out, int out_size, void* d_ws, size_t ws_size,
                              hipStream_t stream) {
    const float* input  = (const float*)d_in[0];   // [8192, 256]
    const float* adj    = (const float*)d_in[1];   // [8192, 8192]
    const float* weight = (const float*)d_in[2];   // [256, 128]
    const float* bias   = (const float*)d_in[3];   // [128]
    float* out = (float*)d_out;                    // [8192, 128]
    float* S   = (float*)d_ws;                     // [8192, 128] = 4 MB scratch

    // S = X @ W
    gcn_wmma_gemm<<<dim3(NODES / 128), dim3(256), 0, stream>>>(
        input, 256, weight, 256, nullptr, S);
    // out = adj @ S + bias
    gcn_wmma_gemm<<<dim3(NODES / 128), dim3(256), 0, stream>>>(
        adj, NODES, S, NODES, bias, out);
}